// GCN_72241349919044
// MI455X (gfx1250) — compile-verified
//
#include <hip/hip_runtime.h>
#include <hip/hip_bf16.h>

typedef __attribute__((ext_vector_type(2))) float v2f;
typedef __attribute__((ext_vector_type(8))) float v8f;

// ---------------- degree kernels ----------------
__global__ void k_fill1(float* p, int n) {
    int i = blockIdx.x * blockDim.x + threadIdx.x;
    if (i < n) p[i] = 1.0f;               // self-loop contributes 1
}
__global__ void k_deg_accum(const int* __restrict__ dst, float* __restrict__ deg, int nE) {
    int e = blockIdx.x * blockDim.x + threadIdx.x;
    if (e < nE) atomicAdd(&deg[dst[e]], 1.0f);
}
__global__ void k_rsqrt_inplace(float* p, int n) {
    int i = blockIdx.x * blockDim.x + threadIdx.x;
    if (i < n) p[i] = 1.0f / sqrtf(p[i]);
}

// ---------------- WMMA GEMM: H[N,64] = A[N,64] * W[64,64] (f32 exact) ----------------
// One wave computes a 16-row x 64-col strip: 4 accumulators, K accumulated 4 at a time.
__global__ __launch_bounds__(128) void k_gemm64_wmma(const float* __restrict__ A,
                                                     const float* __restrict__ W,
                                                     float* __restrict__ H, int nrows) {
    int wave = blockIdx.x * (blockDim.x >> 5) + (threadIdx.x >> 5);
    int lane = threadIdx.x & 31;
    int row0 = wave * 16;
    if (row0 >= nrows) return;            // uniform per wave; EXEC stays all-ones
    int half = lane >> 4;                 // 0: K offsets {0,1}; 1: K offsets {2,3}
    int l16  = lane & 15;

    v8f acc0 = {}, acc1 = {}, acc2 = {}, acc3 = {};
    const float* arow = A + (size_t)(row0 + l16) * 64;

    for (int kb = 0; kb < 64; kb += 4) {
        int k = kb + 2 * half;
        // A fragment (16x4): VGPR v, lane -> A[M=l16][kb + 2*half + v]
        v2f a; a.x = arow[k]; a.y = arow[k + 1];
        // B fragments (4x16): VGPR v, lane -> W[kb + 2*half + v][colTile*16 + l16]
        const float* wk0 = W + (size_t)k * 64;
        const float* wk1 = wk0 + 64;
        v2f b0; b0.x = wk0[l16];      b0.y = wk1[l16];
        v2f b1; b1.x = wk0[16 + l16]; b1.y = wk1[16 + l16];
        v2f b2; b2.x = wk0[32 + l16]; b2.y = wk1[32 + l16];
        v2f b3; b3.x = wk0[48 + l16]; b3.y = wk1[48 + l16];
        acc0 = __builtin_amdgcn_wmma_f32_16x16x4_f32(false, a, false, b0, (short)0, acc0, false, false);
        acc1 = __builtin_amdgcn_wmma_f32_16x16x4_f32(false, a, false, b1, (short)0, acc1, false, false);
        acc2 = __builtin_amdgcn_wmma_f32_16x16x4_f32(false, a, false, b2, (short)0, acc2, false, false);
        acc3 = __builtin_amdgcn_wmma_f32_16x16x4_f32(false, a, false, b3, (short)0, acc3, false, false);
    }
    // C/D layout: VGPR v, lanes 0-15 -> M=v; lanes 16-31 -> M=8+v; N = l16 (+16*tile)
#pragma unroll
    for (int v = 0; v < 8; ++v) {
        float* orow = H + (size_t)(row0 + 8 * half + v) * 64 + l16;
        orow[0]  = acc0[v];
        orow[16] = acc1[v];
        orow[32] = acc2[v];
        orow[48] = acc3[v];
    }
}

// ---------------- GCN aggregation ----------------
// agg[i,:] = h[i,:] * dinv[i]^2   (self-loop term; also initializes agg)
__global__ void k_init_agg(const float* __restrict__ h, const float* __restrict__ dinv,
                           float* __restrict__ agg, int total) {
    int idx = blockIdx.x * blockDim.x + threadIdx.x;
    if (idx >= total) return;
    float di = dinv[idx >> 6];
    agg[idx] = h[idx] * di * di;
}

// One 64-thread group per edge: coalesced gather + coalesced atomic scatter.
__global__ __launch_bounds__(256) void k_scatter(const int* __restrict__ src,
                                                 const int* __restrict__ dst,
                                                 const float* __restrict__ dinv,
                                                 const float* __restrict__ h,
                                                 float* __restrict__ agg, int nE) {
    int e = blockIdx.x * 4 + (threadIdx.x >> 6);
    int c = threadIdx.x & 63;
    if (e >= nE) return;
    int s = src[e], d = dst[e];
    float w = dinv[s] * dinv[d];
    atomicAdd(&agg[(size_t)d * 64 + c], h[(size_t)s * 64 + c] * w);
}

// ---------------- BatchNorm (training mode) ----------------
__global__ void k_zero_stats(float* stats) {
    int t = threadIdx.x;
    if (t < 128) stats[t] = 0.0f;         // sum[64], sumsq[64]
}
__global__ __launch_bounds__(256) void k_stats(const float* __restrict__ agg,
                                               float* __restrict__ stats, int n) {
    __shared__ float ssum[256], ssq[256];
    int c = threadIdx.x & 63;
    int r = threadIdx.x >> 6;             // 4 row-replicas per block
    float sum = 0.f, sq = 0.f;
    for (int i = blockIdx.x * 4 + r; i < n; i += gridDim.x * 4) {
        float v = agg[(size_t)i * 64 + c];
        sum += v; sq += v * v;
    }
    ssum[threadIdx.x] = sum; ssq[threadIdx.x] = sq;
    __syncthreads();
    if (r == 0) {
        sum = ssum[c] + ssum[c + 64] + ssum[c + 128] + ssum[c + 192];
        sq  = ssq [c] + ssq [c + 64] + ssq [c + 128] + ssq [c + 192];
        atomicAdd(&stats[c], sum);
        atomicAdd(&stats[64 + c], sq);
    }
}
// scale = g*rsqrt(var+eps); shift = beta - scale*mean  (pre-BN bias cancels)
__global__ void k_bn_finalize(float* __restrict__ stats, const float* __restrict__ g,
                              const float* __restrict__ beta, float invN) {
    int c = threadIdx.x;                  // 64 threads
    float mu  = stats[c] * invN;
    float var = stats[64 + c] * invN - mu * mu;
    float sc  = g[c] / sqrtf(var + 1e-5f);
    stats[128 + c] = sc;
    stats[192 + c] = beta[c] - sc * mu;
}
__global__ void k_bn_relu(const float* __restrict__ agg, const float* __restrict__ stats,
                          float* __restrict__ act, int total) {
    int idx = blockIdx.x * blockDim.x + threadIdx.x;
    if (idx >= total) return;
    int c = idx & 63;
    float v = stats[128 + c] * agg[idx] + stats[192 + c];
    act[idx] = v > 0.f ? v : 0.f;
}

// ---------------- Output layer (H -> 1) ----------------
__global__ void k_gemv_out(const float* __restrict__ act, const float* __restrict__ Wo,
                           float* __restrict__ ho, int n) {
    int i = blockIdx.x * blockDim.x + threadIdx.x;
    if (i >= n) return;
    const float4* a = (const float4*)(act + (size_t)i * 64);
    const float4* w = (const float4*)Wo;
    float s = 0.f;
#pragma unroll
    for (int k = 0; k < 16; ++k) {
        float4 av = a[k], wv = w[k];
        s += av.x * wv.x + av.y * wv.y + av.z * wv.z + av.w * wv.w;
    }
    ho[i] = s;
}
__global__ void k_init_aggo(const float* __restrict__ ho, const float* __restrict__ dinv,
                            float* __restrict__ aggo, int n) {
    int i = blockIdx.x * blockDim.x + threadIdx.x;
    if (i >= n) return;
    float di = dinv[i];
    aggo[i] = ho[i] * di * di;
}
__global__ void k_scatter_o(const int* __restrict__ src, const int* __restrict__ dst,
                            const float* __restrict__ dinv, const float* __restrict__ ho,
                            float* __restrict__ aggo, int nE) {
    int e = blockIdx.x * blockDim.x + threadIdx.x;
    if (e >= nE) return;
    int s = src[e], d = dst[e];
    atomicAdd(&aggo[d], ho[s] * dinv[s] * dinv[d]);
}
__global__ void k_add_bias(const float* __restrict__ aggo, const float* __restrict__ bo,
                           float* __restrict__ out, int n) {
    int i = blockIdx.x * blockDim.x + threadIdx.x;
    if (i < n) out[i] = aggo[i] + bo[0];
}

// ---------------- launcher ----------------
extern "C" void kernel_launch(void* const* d_in, const int* in_sizes, int n_in,
                              void* d_out, int out_size, void* d_ws, size_t ws_size,
                              hipStream_t stream) {
    const float* x   = (const float*)d_in[0];
    const int*   ei  = (const int*)  d_in[1];
    const float* W1  = (const float*)d_in[2];
    const float* g1  = (const float*)d_in[4];
    const float* be1 = (const float*)d_in[5];
    const float* Wh  = (const float*)d_in[6];
    const float* gh  = (const float*)d_in[8];
    const float* beh = (const float*)d_in[9];
    const float* Wo  = (const float*)d_in[10];
    const float* bo  = (const float*)d_in[11];

    const int N = in_sizes[0] / 64;       // 50000
    const int E = in_sizes[1] / 2;        // 800000
    const int* src = ei;
    const int* dst = ei + E;

    float* wsf   = (float*)d_ws;
    float* dinv  = wsf;                         // N
    float* h     = dinv + N;                    // N*64
    float* agg   = h    + (size_t)N * 64;       // N*64
    float* act   = agg  + (size_t)N * 64;       // N*64
    float* stats = act  + (size_t)N * 64;       // 256
    float* ho    = stats + 256;                 // N
    float* aggo  = ho + N;                      // N

    const int T = 256;
    const int total = N * 64;

    // degrees -> dinv
    k_fill1        <<<(N + T - 1) / T, T, 0, stream>>>(dinv, N);
    k_deg_accum    <<<(E + T - 1) / T, T, 0, stream>>>(dst, dinv, E);
    k_rsqrt_inplace<<<(N + T - 1) / T, T, 0, stream>>>(dinv, N);

    const int tiles = (N + 15) / 16;            // 3125 row tiles
    const int gemmGrid = (tiles + 3) / 4;       // 4 waves / block

    // --- layer 1 ---
    k_gemm64_wmma<<<gemmGrid, 128, 0, stream>>>(x, W1, h, N);
    k_init_agg   <<<(total + T - 1) / T, T, 0, stream>>>(h, dinv, agg, total);
    k_zero_stats <<<1, 128, 0, stream>>>(stats);
    k_scatter    <<<(E + 3) / 4, 256, 0, stream>>>(src, dst, dinv, h, agg, E);
    k_stats      <<<256, 256, 0, stream>>>(agg, stats, N);
    k_bn_finalize<<<1, 64, 0, stream>>>(stats, g1, be1, 1.0f / (float)N);
    k_bn_relu    <<<(total + T - 1) / T, T, 0, stream>>>(agg, stats, act, total);

    // --- layer 2 ---
    k_gemm64_wmma<<<gemmGrid, 128, 0, stream>>>(act, Wh, h, N);
    k_init_agg   <<<(total + T - 1) / T, T, 0, stream>>>(h, dinv, agg, total);
    k_zero_stats <<<1, 128, 0, stream>>>(stats);
    k_scatter    <<<(E + 3) / 4, 256, 0, stream>>>(src, dst, dinv, h, agg, E);
    k_stats      <<<256, 256, 0, stream>>>(agg, stats, N);
    k_bn_finalize<<<1, 64, 0, stream>>>(stats, gh, beh, 1.0f / (float)N);
    k_bn_relu    <<<(total + T - 1) / T, T, 0, stream>>>(agg, stats, act, total);

    // --- output layer (H -> 1) ---
    k_gemv_out  <<<(N + T - 1) / T, T, 0, stream>>>(act, Wo, ho, N);
    k_init_aggo <<<(N + T - 1) / T, T, 0, stream>>>(ho, dinv, aggo, N);
    k_scatter_o <<<(E + T - 1) / T, T, 0, stream>>>(src, dst, dinv, ho, aggo, E);
    k_add_bias  <<<(N + T - 1) / T, T, 0, stream>>>(aggo, bo, (float*)d_out, N);
}